// Encoder_53412213293256
// MI455X (gfx1250) — compile-verified
//
#include <hip/hip_runtime.h>

#define DHC 128
#define EPSC 1e-5f

typedef __attribute__((ext_vector_type(2))) float v2f;
typedef __attribute__((ext_vector_type(8))) float v8f;

__device__ __forceinline__ void atomAddF(float* p, float v) {
  __hip_atomic_fetch_add(p, v, __ATOMIC_RELAXED, __HIP_MEMORY_SCOPE_AGENT);
}

// ---------------- utility kernels ----------------
__global__ void zero_u32(unsigned* __restrict__ p, int n) {
  int i = blockIdx.x * 256 + threadIdx.x;
  if (i < n) p[i] = 0u;
}
__global__ void zero_f32(float* __restrict__ p, int n) {
  int i = blockIdx.x * 256 + threadIdx.x;
  if (i < n) p[i] = 0.0f;
}

// count degrees for all 3 edge types in one pass: cnt[e*N + idx[i]] += 1
__global__ void degree_count(const int* __restrict__ idx, unsigned* __restrict__ cnt,
                             int nedges_per_type, int nnodes) {
  int i = blockIdx.x * 256 + threadIdx.x;
  int total = 3 * nedges_per_type;
  if (i >= total) return;
  int e = i / nedges_per_type;
  __hip_atomic_fetch_add(&cnt[(size_t)e * nnodes + idx[i]], 1u,
                         __ATOMIC_RELAXED, __HIP_MEMORY_SCOPE_AGENT);
}

__global__ void degree_rsqrt(const unsigned* __restrict__ cnt, float* __restrict__ rsq, int n) {
  int i = blockIdx.x * 256 + threadIdx.x;
  if (i >= n) return;
  float d = (float)cnt[i];
  rsq[i] = rsqrtf(d < 1.0f ? 1.0f : d);
}

// acc[n][c] = b[0][c] + b[1][c] + b[2][c]  (sum of the 3 edge-type biases)
__global__ void init_acc(const float* __restrict__ b3, float* __restrict__ acc, long long total) {
  long long i = (long long)blockIdx.x * 256 + threadIdx.x;
  if (i >= total) return;
  int c = (int)(i & (DHC - 1));
  acc[i] = b3[c] + b3[DHC + c] + b3[2 * DHC + c];
}

// ---------------- fp32 WMMA GEMM: Y[r][:] = (scale[r]*X[r][:]) @ W (+bias, relu) ----------------
// X: [nrows,128], W: [128,128] row-major (W[k][n]), Y: [nrows,128]
// Block = 256 threads = 8 waves; each wave owns 16 rows x 128 cols (8 WMMA tiles).
// W staged in LDS K-pair-interleaved: sW[(k>>1)*256 + n*2 + (k&1)], so each B
// fragment {W[k][n], W[k+1][n]} is one 8-byte ds_load_b64 into the operand pair.
__global__ __launch_bounds__(256)
void gemm128_wmma(const float* __restrict__ X, const float* __restrict__ scale,
                  const float* __restrict__ W, const float* __restrict__ bias,
                  float* __restrict__ Y, int nrows, int dorelu) {
  __shared__ float sW[DHC * DHC];  // 64 KB
  const int tid = threadIdx.x;
  const float4* W4 = (const float4*)W;
#pragma unroll
  for (int i = 0; i < 16; ++i) {
    int f4 = tid + i * 256;        // float4 index 0..4095
    float4 v = W4[f4];
    int k = f4 >> 5;               // source row
    int n = (f4 & 31) * 4;         // source col base
    float* dstp = &sW[(k >> 1) * 256 + (k & 1)];
    dstp[(n + 0) * 2] = v.x;
    dstp[(n + 1) * 2] = v.y;
    dstp[(n + 2) * 2] = v.z;
    dstp[(n + 3) * 2] = v.w;
  }
  __syncthreads();

  const int wave = tid >> 5;
  const int lane = tid & 31;
  const int hs = lane >> 4;    // 0: lanes 0-15, 1: lanes 16-31
  const int l15 = lane & 15;

  // A fragment source row for this lane (16x4 f32 layout: lanes 0-15 -> M=0..15)
  const int rowA = blockIdx.x * 128 + wave * 16 + l15;
  const int rA = rowA < nrows ? rowA : (nrows - 1);  // clamp loads; stores guarded
  const float sc = scale ? scale[rA] : 1.0f;
  const float* xrow = X + (size_t)rA * DHC;
  const float* sWp = &sW[l15 * 2];

  v8f acc[8];
#pragma unroll
  for (int t = 0; t < 8; ++t)
#pragma unroll
    for (int v = 0; v < 8; ++v) acc[t][v] = 0.0f;

  for (int kk = 0; kk < DHC; kk += 4) {
    const int ak = kk + hs * 2;  // lanes 0-15 carry K=kk,kk+1 ; lanes 16-31 carry K=kk+2,kk+3
    v2f a;
    a.x = xrow[ak] * sc;
    a.y = xrow[ak + 1] * sc;
    const int krow = (ak >> 1) * 256;  // ak is even
#pragma unroll
    for (int t = 0; t < 8; ++t) {
      v2f b = *(const v2f*)(sWp + krow + t * 32);  // {W[ak][col], W[ak+1][col]}
      acc[t] = __builtin_amdgcn_wmma_f32_16x16x4_f32(
          false, a, false, b, (short)0, acc[t], false, false);
    }
  }

  // C/D layout: VGPR v, lanes 0-15 -> M=v ; lanes 16-31 -> M=v+8 ; N = l15
  const int rowBase = blockIdx.x * 128 + wave * 16 + hs * 8;
#pragma unroll
  for (int t = 0; t < 8; ++t) {
    const int col = t * 16 + l15;
    const float bb = bias ? bias[col] : 0.0f;
#pragma unroll
    for (int v = 0; v < 8; ++v) {
      const int r = rowBase + v;
      if (r < nrows) {
        float val = acc[t][v] + bb;
        if (dorelu) val = fmaxf(val, 0.0f);
        Y[(size_t)r * DHC + col] = val;
      }
    }
  }
}

// ---------------- edge scatter: acc[dst] += rin[dst] * he[src]  (32 threads / edge) ----------------
__global__ void scatter_edges(const float* __restrict__ he, const int* __restrict__ src,
                              const int* __restrict__ dst, const float* __restrict__ rin,
                              float* __restrict__ acc, int nedges) {
  long long gid = (long long)blockIdx.x * 256 + threadIdx.x;
  int e = (int)(gid >> 5);
  if (e >= nedges) return;
  int c = ((int)gid & 31) * 4;
  int s = src[e];
  int d = dst[e];
  float r = rin[d];
  float4 v = *(const float4*)(he + (size_t)s * DHC + c);
  float* ap = acc + (size_t)d * DHC + c;
  atomAddF(ap + 0, v.x * r);
  atomAddF(ap + 1, v.y * r);
  atomAddF(ap + 2, v.z * r);
  atomAddF(ap + 3, v.w * r);
}

// ---------------- batchnorm ----------------
__global__ void bn_reduce(const float* __restrict__ xin, float* __restrict__ scratch, int nrows) {
  int c = threadIdx.x & (DHC - 1);
  int g = blockIdx.x * 2 + (threadIdx.x >> 7);
  float s = 0.0f, s2 = 0.0f;
  for (int r = g; r < nrows; r += (int)gridDim.x * 2) {
    float v = xin[(size_t)r * DHC + c];
    s += v;
    s2 += v * v;
  }
  atomAddF(&scratch[c], s);
  atomAddF(&scratch[DHC + c], s2);
}

__global__ void bn_finalize(float* __restrict__ scratch, float inv_n) {
  int c = threadIdx.x;
  if (c < DHC) {
    float mu = scratch[c] * inv_n;
    float var = scratch[DHC + c] * inv_n - mu * mu;
    scratch[c] = mu;
    scratch[DHC + c] = rsqrtf(var + EPSC);
  }
}

__global__ void bn_apply(const float* __restrict__ xin, const float* __restrict__ scratch,
                         const float* __restrict__ gamma, const float* __restrict__ beta,
                         float* __restrict__ out, long long total) {
  long long i = (long long)blockIdx.x * 256 + threadIdx.x;
  if (i >= total) return;
  int c = (int)(i & (DHC - 1));
  out[i] = (xin[i] - scratch[c]) * scratch[DHC + c] * gamma[c] + beta[c];
}

// ---------------- host ----------------
extern "C" void kernel_launch(void* const* d_in, const int* in_sizes, int n_in,
                              void* d_out, int out_size, void* d_ws, size_t ws_size,
                              hipStream_t stream) {
  (void)n_in; (void)out_size; (void)ws_size;
  const float* x     = (const float*)d_in[0];
  const int*   src   = (const int*)d_in[1];
  const int*   dst   = (const int*)d_in[2];
  const float* gc_W0 = (const float*)d_in[3];
  const float* gc_b0 = (const float*)d_in[4];
  const float* gc_W  = (const float*)d_in[5];
  const float* gc_b  = (const float*)d_in[6];
  const float* fc_W  = (const float*)d_in[7];
  const float* fc_b  = (const float*)d_in[8];
  const float* bn_g  = (const float*)d_in[9];
  const float* bn_b  = (const float*)d_in[10];

  const int Nn = in_sizes[0] / DHC;  // 100000
  const int Ee = in_sizes[1] / 3;    // 1600000

  char* ws = (char*)d_ws;
  unsigned* cnt  = (unsigned*)ws;                                 // 6*Nn u32
  float* rsq     = (float*)(ws + (size_t)6 * Nn * 4);             // 6*Nn f32 (rout[e], rin[e])
  float* h       = (float*)(ws + (size_t)12 * Nn * 4);            // Nn*128
  float* he      = h + (size_t)Nn * DHC;                          // Nn*128
  float* acc     = he + (size_t)Nn * DHC;                         // Nn*128
  float* scratch = acc + (size_t)Nn * DHC;                        // 256 f32

  const int tot6 = 6 * Nn;
  const int tot3e = 3 * Ee;
  zero_u32<<<(tot6 + 255) / 256, 256, 0, stream>>>(cnt, tot6);
  degree_count<<<(tot3e + 255) / 256, 256, 0, stream>>>(src, cnt, Ee, Nn);                  // out-deg
  degree_count<<<(tot3e + 255) / 256, 256, 0, stream>>>(dst, cnt + (size_t)3 * Nn, Ee, Nn); // in-deg
  degree_rsqrt<<<(tot6 + 255) / 256, 256, 0, stream>>>(cnt, rsq, tot6);

  const long long totNC = (long long)Nn * DHC;
  const int gemmBlocks = (Nn + 127) / 128;
  const int scatBlocks = (int)(((long long)Ee * 32 + 255) / 256);
  const int elemBlocks = (int)((totNC + 255) / 256);

  const float* in = x;
  for (int l = 0; l < 3; ++l) {
    const float* Wl = (l == 0) ? gc_W0 : gc_W + (size_t)(l - 1) * 3 * DHC * DHC;
    const float* bl = (l == 0) ? gc_b0 : gc_b + (size_t)(l - 1) * 3 * DHC;

    init_acc<<<elemBlocks, 256, 0, stream>>>(bl, acc, totNC);
    for (int e = 0; e < 3; ++e) {
      gemm128_wmma<<<gemmBlocks, 256, 0, stream>>>(
          in, rsq + (size_t)e * Nn, Wl + (size_t)e * DHC * DHC, nullptr, he, Nn, 0);
      scatter_edges<<<scatBlocks, 256, 0, stream>>>(
          he, src + (size_t)e * Ee, dst + (size_t)e * Ee, rsq + (size_t)(3 + e) * Nn, acc, Ee);
    }
    // FC + bias + ReLU into he (free after scatters)
    gemm128_wmma<<<gemmBlocks, 256, 0, stream>>>(
        acc, nullptr, fc_W + (size_t)l * DHC * DHC, fc_b + (size_t)l * DHC, he, Nn, 1);
    // BatchNorm (training mode, biased stats)
    zero_f32<<<1, 256, 0, stream>>>(scratch, 2 * DHC);
    bn_reduce<<<256, 256, 0, stream>>>(he, scratch, Nn);
    bn_finalize<<<1, 128, 0, stream>>>(scratch, 1.0f / (float)Nn);
    float* outp = (l == 2) ? (float*)d_out : h;
    bn_apply<<<elemBlocks, 256, 0, stream>>>(
        he, scratch, bn_g + (size_t)l * DHC, bn_b + (size_t)l * DHC, outp, totNC);
    in = h;
  }
}